// MAMDense_31988916420623
// MI455X (gfx1250) — compile-verified
//
#include <hip/hip_runtime.h>
#include <math.h>

// ---- Tiling config -------------------------------------------------------
#define BM 64           // C-tile rows per block
#define BN 64           // C-tile cols per block
#define BK 32           // k-chunk
#define LDA 68          // BM + 4 pad floats -> 272B rows (16B aligned, bank-spread)
#define TM 4            // rows per thread
#define TN 4            // cols per thread
#define NTHREADS 256    // 16x16 threads = 8 wave32

typedef __attribute__((address_space(1))) int as1_int;
typedef __attribute__((address_space(3))) int as3_int;
typedef __attribute__((address_space(3))) char as3_char;

// ---- gfx1250 async global->LDS copy (per-lane LDS scatter => free transpose)
__device__ __forceinline__ void async_copy_b32(const float* gsrc, float* ldst) {
#if __has_builtin(__builtin_amdgcn_global_load_async_to_lds_b32)
  __builtin_amdgcn_global_load_async_to_lds_b32(
      (as1_int*)gsrc, (as3_int*)ldst, /*imm offset*/0, /*cpol*/0);
#else
  unsigned laddr = (unsigned)(unsigned long long)(as3_char*)ldst;
  asm volatile("global_load_async_to_lds_b32 %0, %1, off"
               :: "v"(laddr), "v"(gsrc) : "memory");
#endif
}

__device__ __forceinline__ void wait_async_le16() {
#if __has_builtin(__builtin_amdgcn_s_wait_asynccnt)
  __builtin_amdgcn_s_wait_asynccnt(16);
#else
  asm volatile("s_wait_asynccnt 16" ::: "memory");
#endif
}

__device__ __forceinline__ void wait_async_le0() {
#if __has_builtin(__builtin_amdgcn_s_wait_asynccnt)
  __builtin_amdgcn_s_wait_asynccnt(0);
#else
  asm volatile("s_wait_asynccnt 0" ::: "memory");
#endif
}

// ---- MAM dense kernel ----------------------------------------------------
// C[m,n] = max_k(A[m,k]*W[n,k]) + min_k(A[m,k]*W[n,k]) + bias[n]
// plus argmax / argmin planes (first-occurrence ties, matching jnp.argmax).
__global__ __launch_bounds__(NTHREADS)
void MAMDense_31988916420623_kernel(const float* __restrict__ A,    // [M,K]
                                    const float* __restrict__ W,    // [N,K]
                                    const float* __restrict__ bias, // [N]
                                    float* __restrict__ outC,       // [M,N]
                                    int* __restrict__ outAmx,       // [M,N]
                                    int* __restrict__ outAmn,       // [M,N]
                                    int M, int N, int K) {
  __shared__ __align__(16) float ldsA[2][BK][LDA];  // transposed: [k][m]
  __shared__ __align__(16) float ldsW[2][BK][LDA];  // transposed: [k][n]

  const int tid = threadIdx.x;
  const int tx  = tid & 15;    // col group: cols tx*4..tx*4+3
  const int ty  = tid >> 4;    // row group: rows ty*4..ty*4+3
  const int m0  = blockIdx.y * BM;
  const int n0  = blockIdx.x * BN;

  // Async-copy mapping: lane handles fixed k-column kk0, rows r0+8*i.
  // Consecutive lanes -> consecutive k in one global row => coalesced 128B/wave.
  // LDS write ldsA[kk0][r] => on-the-fly transpose via per-lane LDS address.
  const int kk0 = tid & (BK - 1);   // 0..31
  const int r0  = tid >> 5;         // 0..7
  const int NK  = K / BK;

  // Prime the pipeline: chunk 0 into buffer 0.
  {
    const float* ga = A + (size_t)(m0 + r0) * K + kk0;
    const float* gw = W + (size_t)(n0 + r0) * K + kk0;
#pragma unroll
    for (int i = 0; i < 8; ++i) {
      async_copy_b32(ga + (size_t)(8 * i) * K, &ldsA[0][kk0][r0 + 8 * i]);
      async_copy_b32(gw + (size_t)(8 * i) * K, &ldsW[0][kk0][r0 + 8 * i]);
    }
  }

  float maxv[TM][TN], minv[TM][TN];
  int   maxi[TM][TN], mini[TM][TN];
#pragma unroll
  for (int i = 0; i < TM; ++i)
#pragma unroll
    for (int j = 0; j < TN; ++j) {
      maxv[i][j] = -INFINITY;
      minv[i][j] =  INFINITY;
      maxi[i][j] = 0;
      mini[i][j] = 0;
    }

  for (int kc = 0; kc < NK; ++kc) {
    const int buf = kc & 1;

    if (kc + 1 < NK) {
      // Prefetch next chunk into the other buffer, then wait for current
      // chunk (in-order ASYNCcnt: <=16 outstanding == next chunk only).
      const int nb = buf ^ 1;
      const int kb = (kc + 1) * BK;
      const float* ga = A + (size_t)(m0 + r0) * K + kb + kk0;
      const float* gw = W + (size_t)(n0 + r0) * K + kb + kk0;
#pragma unroll
      for (int i = 0; i < 8; ++i) {
        async_copy_b32(ga + (size_t)(8 * i) * K, &ldsA[nb][kk0][r0 + 8 * i]);
        async_copy_b32(gw + (size_t)(8 * i) * K, &ldsW[nb][kk0][r0 + 8 * i]);
      }
      wait_async_le16();
    } else {
      wait_async_le0();
    }
    __syncthreads();   // current chunk visible to all 8 waves

    const float* arow = &ldsA[buf][0][ty * TM];
    const float* wrow = &ldsW[buf][0][tx * TN];
    int kg = kc * BK;
#pragma unroll 4
    for (int kk = 0; kk < BK; ++kk, ++kg) {
      const float4 av = *(const float4*)(arow + kk * LDA);  // ds_load_b128 (bcast)
      const float4 wv = *(const float4*)(wrow + kk * LDA);  // ds_load_b128
      const float a[TM] = {av.x, av.y, av.z, av.w};
      const float w[TN] = {wv.x, wv.y, wv.z, wv.w};
#pragma unroll
      for (int i = 0; i < TM; ++i)
#pragma unroll
        for (int j = 0; j < TN; ++j) {
          const float p = a[i] * w[j];
          if (p > maxv[i][j]) { maxv[i][j] = p; maxi[i][j] = kg; }
          if (p < minv[i][j]) { minv[i][j] = p; mini[i][j] = kg; }
        }
    }
    __syncthreads();   // protect buffer before chunk kc+2 overwrites it
  }

  // Epilogue: vectorized 128-bit stores (n0 + tx*4 is 16B aligned).
  const float4 bv = *(const float4*)(bias + n0 + tx * TN);
  const float b[TN] = {bv.x, bv.y, bv.z, bv.w};
#pragma unroll
  for (int i = 0; i < TM; ++i) {
    const size_t row = (size_t)(m0 + ty * TM + i) * N + (n0 + tx * TN);
    float4 c;
    c.x = maxv[i][0] + minv[i][0] + b[0];
    c.y = maxv[i][1] + minv[i][1] + b[1];
    c.z = maxv[i][2] + minv[i][2] + b[2];
    c.w = maxv[i][3] + minv[i][3] + b[3];
    int4 mx = {maxi[i][0], maxi[i][1], maxi[i][2], maxi[i][3]};
    int4 mn = {mini[i][0], mini[i][1], mini[i][2], mini[i][3]};
    *(float4*)(outC + row)  = c;
    *(int4*)(outAmx + row)  = mx;
    *(int4*)(outAmn + row)  = mn;
  }
}

// ---- Launch --------------------------------------------------------------
extern "C" void kernel_launch(void* const* d_in, const int* in_sizes, int n_in,
                              void* d_out, int out_size, void* d_ws, size_t ws_size,
                              hipStream_t stream) {
  (void)n_in; (void)out_size; (void)d_ws; (void)ws_size;
  const float* x    = (const float*)d_in[0];   // [B,S,K] flat = [M,K]
  const float* wgt  = (const float*)d_in[1];   // [N,K]
  const float* bias = (const float*)d_in[2];   // [N]

  const int N = in_sizes[2];                   // 1024
  const int K = in_sizes[1] / N;               // 1024
  const int M = in_sizes[0] / K;               // 8192

  float* outC  = (float*)d_out;
  int*   amx   = (int*)d_out + (size_t)M * N;
  int*   amn   = (int*)d_out + 2 * (size_t)M * N;

  dim3 grid(N / BN, M / BM);                   // (16, 128)
  MAMDense_31988916420623_kernel<<<grid, NTHREADS, 0, stream>>>(
      x, wgt, bias, outC, amx, amn, M, N, K);
}